// OhemMaskedBCEWithLogitsLoss_4440996184820
// MI455X (gfx1250) — compile-verified
//
#include <hip/hip_runtime.h>
#include <hip/hip_bf16.h>
#include <stdint.h>

#define C               64
#define HEPC            15
#define TPB             256
#define ROWS_PER_BLOCK  512
#define TILE_ROWS       64
#define NTILES          (ROWS_PER_BLOCK / TILE_ROWS)   // 8
#define TILE_BYTES      (TILE_ROWS * C * 4)            // 16 KB
#define EPSV            1e-8f
#define NEG_INF         (-__builtin_inff())

// ---- CDNA5 async global->LDS copy (ASYNCcnt path, gfx1250) -----------------
// global_load_async_to_lds_b128 vdst(lds byte offset), vaddr(32b offset), saddr
__device__ __forceinline__ void async_load_b128(uint32_t lds_off, uint32_t voff,
                                                const void* sbase) {
  asm volatile("global_load_async_to_lds_b128 %0, %1, %2"
               :: "v"(lds_off), "v"(voff), "s"(sbase)
               : "memory");
}

template <int N>
__device__ __forceinline__ void wait_asynccnt() {
#if __has_builtin(__builtin_amdgcn_s_wait_asynccnt)
  __builtin_amdgcn_s_wait_asynccnt(N);
#else
  asm volatile("s_wait_asynccnt %0" :: "i"(N) : "memory");
#endif
}

// ---- register-resident exact top-15 (ascending; best[0] = current min) -----
__device__ __forceinline__ void topk_insert(float (&best)[HEPC], float v) {
  if (v > best[0]) {
    best[0] = v;
#pragma unroll
    for (int j = 0; j < HEPC - 1; ++j) {       // one bubble pass re-sorts
      float a = best[j], b = best[j + 1];
      best[j]     = fminf(a, b);
      best[j + 1] = fmaxf(a, b);
    }
  }
}

// ============================ stage 1 =======================================
__global__ void __launch_bounds__(TPB)
ohem_stage1(const float* __restrict__ logits, const float* __restrict__ target,
            float* __restrict__ cand) {
  // [buf0: L|T][buf1: L|T] = 64 KB; merge scratch aliases after drain
  __shared__ __align__(16) unsigned char smem[4 * TILE_BYTES];

  const int tid = threadIdx.x;
  const int c   = tid & (C - 1);   // class column
  const int sub = tid >> 6;        // 0..3 row sub-lane group
  const uint32_t smem_base = (uint32_t)(uintptr_t)(void*)smem;  // LDS offset
  const long long blockRow0 = (long long)blockIdx.x * ROWS_PER_BLOCK;
  const uint32_t voff0 = (uint32_t)tid * 16u;

  float best[HEPC];
#pragma unroll
  for (int j = 0; j < HEPC; ++j) best[j] = NEG_INF;

  auto issue = [&](int tile, int buf) {
    const char* gL = (const char*)(logits + (blockRow0 + (long long)tile * TILE_ROWS) * C);
    const char* gT = (const char*)(target + (blockRow0 + (long long)tile * TILE_ROWS) * C);
    const uint32_t ldsL = smem_base + (uint32_t)buf * (2u * TILE_BYTES);
    const uint32_t ldsT = ldsL + TILE_BYTES;
#pragma unroll
    for (int k = 0; k < 4; ++k) {              // 256 thr * 16 B * 4 = 16 KB
      const uint32_t o = voff0 + (uint32_t)k * 4096u;
      async_load_b128(ldsL + o, o, gL);
      async_load_b128(ldsT + o, o, gT);
    }                                          // 8 async ops / thread / tile
  };

  issue(0, 0);
  for (int i = 0; i < NTILES; ++i) {
    const int nxt = (i + 1 < NTILES) ? (i + 1) : i;  // last iter: dummy refill
    issue(nxt, (i + 1) & 1);
    wait_asynccnt<8>();          // in-order: tile i's 8 ops have completed
    __syncthreads();             // all waves' portions visible in LDS
    const float* sL = (const float*)(smem + (i & 1) * (2 * TILE_BYTES));
    const float* sT = (const float*)(smem + (i & 1) * (2 * TILE_BYTES) + TILE_BYTES);
#pragma unroll
    for (int m = 0; m < TILE_ROWS / 4; ++m) {
      const int r = sub + 4 * m;
      const float x = sL[r * C + c];           // conflict-free: 64 banks
      const float t = sT[r * C + c];
      if (t != -1.0f) {                        // ignore_index
        const float z  = (t > 0.5f) ? x : -x;  // -log(sigmoid(±x)+eps)
        const float s  = 1.0f / (1.0f + __expf(-z));
        const float ce = -__logf(s + EPSV);
        topk_insert(best, ce);
      }
    }
    __syncthreads();             // readers done before buffer is refilled
  }
  wait_asynccnt<0>();            // drain dummy before aliasing smem
  __syncthreads();

  // block-level merge: 4 sub-lists per class -> 15 candidates per class
  float* mf = (float*)smem;                    // 64*4*15 floats = 15 KB
  const int slot = c * 4 + sub;
#pragma unroll
  for (int j = 0; j < HEPC; ++j) mf[slot * HEPC + j] = best[j];
  __syncthreads();
  if (tid < C) {
    float b2[HEPC];
#pragma unroll
    for (int j = 0; j < HEPC; ++j) b2[j] = NEG_INF;
    for (int i2 = 0; i2 < 4 * HEPC; ++i2)
      topk_insert(b2, mf[tid * (4 * HEPC) + i2]);
    float* dst = cand + ((size_t)blockIdx.x * C + tid) * HEPC;
#pragma unroll
    for (int j = 0; j < HEPC; ++j) dst[j] = b2[j];
  }
}

// ============================ stage 2 =======================================
__global__ void __launch_bounds__(256)
ohem_stage2(const float* __restrict__ cand, float* __restrict__ cls_mean, int nb) {
  __shared__ float s[256 * HEPC];
  const int c = blockIdx.x, tid = threadIdx.x;

  float best[HEPC];
#pragma unroll
  for (int j = 0; j < HEPC; ++j) best[j] = NEG_INF;
  for (int b = tid; b < nb; b += 256)
#pragma unroll
    for (int j = 0; j < HEPC; ++j)
      topk_insert(best, cand[((size_t)b * C + c) * HEPC + j]);
#pragma unroll
  for (int j = 0; j < HEPC; ++j) s[tid * HEPC + j] = best[j];
  __syncthreads();

  float b2[HEPC];
#pragma unroll
  for (int j = 0; j < HEPC; ++j) b2[j] = NEG_INF;
  if (tid < 16)
    for (int i = 0; i < 16 * HEPC; ++i)
      topk_insert(b2, s[tid * (16 * HEPC) + i]);
  __syncthreads();
  if (tid < 16)
#pragma unroll
    for (int j = 0; j < HEPC; ++j) s[tid * HEPC + j] = b2[j];
  __syncthreads();

  if (tid == 0) {
    float b3[HEPC];
#pragma unroll
    for (int j = 0; j < HEPC; ++j) b3[j] = NEG_INF;
    for (int i = 0; i < 16 * HEPC; ++i) topk_insert(b3, s[i]);
    float sum = 0.0f;
#pragma unroll
    for (int j = 0; j < HEPC; ++j) sum += b3[j];
    cls_mean[c] = sum * (1.0f / HEPC);
  }
}

// ============================ stage 3 =======================================
__global__ void ohem_stage3(const float* __restrict__ cls_mean,
                            float* __restrict__ out) {
  if (threadIdx.x == 0 && blockIdx.x == 0) {
    float sum = 0.0f;
    for (int c2 = 0; c2 < C; ++c2) sum += cls_mean[c2];
    out[0] = sum * (1.0f / C);
  }
}

// ============================ launcher ======================================
extern "C" void kernel_launch(void* const* d_in, const int* in_sizes, int n_in,
                              void* d_out, int out_size, void* d_ws, size_t ws_size,
                              hipStream_t stream) {
  const float* logits = (const float*)d_in[0];
  const float* target = (const float*)d_in[1];
  const int total = in_sizes[0];            // N*C
  const int N  = total / C;                 // 262144
  const int nb = N / ROWS_PER_BLOCK;        // 512

  float* cand     = (float*)d_ws;                         // nb*C*HEPC floats (~2 MB)
  float* cls_mean = cand + (size_t)nb * C * HEPC;         // +64 floats

  ohem_stage1<<<nb, TPB, 0, stream>>>(logits, target, cand);
  ohem_stage2<<<C, 256, 0, stream>>>(cand, cls_mean, nb);
  ohem_stage3<<<1, 64, 0, stream>>>(cls_mean, (float*)d_out);
}